// VQLayer_89481348645608
// MI455X (gfx1250) — compile-verified
//
#include <hip/hip_runtime.h>

// VQ layer for MI455X (gfx1250): fp32 WMMA (V_WMMA_F32_16X16X4_F32) GEMM-argmin.
// Shapes fixed by the reference: input (64,64,32,32) f32, codebook (1024,64) f32.
// d_out = [idx (65536 floats)] ++ [embed (64,64,32,32) f32].

typedef __attribute__((ext_vector_type(2))) float v2f;
typedef __attribute__((ext_vector_type(8))) float v8f;

#define DIM        64
#define KCODES     1024
#define HW         1024          // H*W = 32*32
#define NPTS       65536         // B*H*W
#define LDS_STRIDE 68            // 64 + 4 pad: bank = (68*l+d)%64 = (4l+d)%64 -> conflict-free, float4-aligned
#define WAVES_PB   16
#define THREADS    (WAVES_PB * 32)

__global__ __launch_bounds__(THREADS)
void vq_wmma_kernel(const float* __restrict__ in, const float* __restrict__ cb,
                    float* __restrict__ out_idx, float* __restrict__ out_emb)
{
    __shared__ float ldsCB[KCODES * LDS_STRIDE];  // 278528 B
    __shared__ float ldsCN[KCODES];               //   4096 B
    __shared__ int   ldsIdx[WAVES_PB * 16];       //   1024 B

    const int tid = threadIdx.x;

    // Stage full codebook into LDS (row stride 68), float4-vectorized, coalesced.
    for (int i = tid; i < KCODES * (DIM / 4); i += THREADS) {
        const int n  = i >> 4;
        const int dq = (i & 15) << 2;
        const float4 v = *(const float4*)(cb + n * DIM + dq);
        *(float4*)(&ldsCB[n * LDS_STRIDE + dq]) = v;
    }
    // Codebook squared norms.
    for (int n = tid; n < KCODES; n += THREADS) {
        const float* r = cb + n * DIM;
        float s = 0.f;
        #pragma unroll
        for (int d = 0; d < DIM; ++d) s += r[d] * r[d];
        ldsCN[n] = s;
    }
    __syncthreads();

    const int lane = tid & 31;
    const int wave = tid >> 5;
    const int half = lane >> 4;   // 0: lanes 0-15, 1: lanes 16-31
    const int l16  = lane & 15;

    const int tile = blockIdx.x * WAVES_PB + wave;   // 16-point row tile per wave
    const int n0   = tile * 16;
    const int bb   = n0 / HW;
    const int hw0  = n0 % HW;
    const float* base = in + (size_t)bb * (DIM * HW) + hw0;

    // A tile (16x64 f32) in WMMA 16x16x4 A layout: per K-chunk k, lane holds
    // features {4k+2*half, 4k+2*half+1} of point l16. Coalesced 64B loads.
    v2f a[16];
    #pragma unroll
    for (int k = 0; k < 16; ++k) {
        const int d0 = k * 4 + 2 * half;
        a[k].x = base[(size_t)d0 * HW + l16];
        a[k].y = base[(size_t)(d0 + 1) * HW + l16];
    }

    float best[8];
    int   bn[8];
    #pragma unroll
    for (int r = 0; r < 8; ++r) { best[r] = 3.4e38f; bn[r] = 0; }

    // Sweep the 64 code tiles; argmin target = ||c||^2 - 2 x.c  (||x||^2 dropped).
    for (int t = 0; t < KCODES / 16; ++t) {
        v8f acc = {0.f, 0.f, 0.f, 0.f, 0.f, 0.f, 0.f, 0.f};
        const int cbase = (t * 16 + l16) * LDS_STRIDE + 2 * half;
        #pragma unroll
        for (int k = 0; k < 16; ++k) {
            const v2f bv = *(const v2f*)(&ldsCB[cbase + k * 4]);   // conflict-free ds_load_b64
            acc = __builtin_amdgcn_wmma_f32_16x16x4_f32(
                false, a[k], false, bv, (short)0, acc, false, false);
        }
        const float cn   = ldsCN[t * 16 + l16];
        const int   cand = t * 16 + l16;   // this lane's code column
        #pragma unroll
        for (int r = 0; r < 8; ++r) {
            const float s = cn - 2.0f * acc[r];
            if (s < best[r]) { best[r] = s; bn[r] = cand; }   // strict < keeps earliest index
        }
    }

    // Per-row argmin: reduce across the 16 lanes holding the 16 columns.
    // Tie-break toward smaller index to match jnp.argmin (first occurrence).
    #pragma unroll
    for (int r = 0; r < 8; ++r) {
        #pragma unroll
        for (int m = 8; m >= 1; m >>= 1) {
            const float os = __shfl_xor(best[r], m, 32);
            const int   on = __shfl_xor(bn[r],   m, 32);
            if (os < best[r] || (os == best[r] && on < bn[r])) { best[r] = os; bn[r] = on; }
        }
        // VGPR r: lanes 0-15 -> row r, lanes 16-31 -> row r+8
        if (l16 == 0) ldsIdx[wave * 16 + r + 8 * half] = bn[r];
    }
    __syncthreads();

    // Index output (float-encoded, exact for idx < 2^24).
    if (lane < 16) out_idx[n0 + lane] = (float)ldsIdx[wave * 16 + lane];

    // Embed gather/scatter: lane owns row l16, feature half `half`.
    // Feature-major writes -> 16 consecutive points per store group = coalesced.
    const int myidx = ldsIdx[wave * 16 + l16];
    float* ob = out_emb + (size_t)bb * (DIM * HW) + hw0 + l16;
    #pragma unroll
    for (int j = 0; j < 32; ++j) {
        const int d = half * 32 + j;
        ob[(size_t)d * HW] = ldsCB[myidx * LDS_STRIDE + d];
    }
}

extern "C" void kernel_launch(void* const* d_in, const int* in_sizes, int n_in,
                              void* d_out, int out_size, void* d_ws, size_t ws_size,
                              hipStream_t stream) {
    const float* in = (const float*)d_in[0];   // (64,64,32,32) f32
    const float* cb = (const float*)d_in[1];   // (1024,64) f32
    float* out_idx = (float*)d_out;            // first 65536 = indices
    float* out_emb = out_idx + NPTS;           // then embeddings

    const int blocks = NPTS / (16 * WAVES_PB); // 256
    vq_wmma_kernel<<<blocks, THREADS, 0, stream>>>(in, cb, out_idx, out_emb);
}